// AttentionMemorySystem_70068096467161
// MI455X (gfx1250) — compile-verified
//
#include <hip/hip_runtime.h>

typedef float v4f __attribute__((ext_vector_type(4)));
typedef float v2f __attribute__((ext_vector_type(2)));
typedef float v8f __attribute__((ext_vector_type(8)));

// ---------------------------------------------------------------------------
// Kernel A: new_mem = memory_attentions with rows [mi, mi+B) mod M <- features
// Pure streaming copy, 16B (b128) per thread, non-temporal both directions.
// Branch-free circular test: i = (row - mi) mod M  via one compare-subtract.
// ---------------------------------------------------------------------------
__global__ void mem_copy_kernel(const float* __restrict__ features,
                                const float* __restrict__ mem,
                                const int* __restrict__ mem_index,
                                float* __restrict__ out,
                                unsigned M, unsigned B, unsigned d4shift) {
    unsigned v    = blockIdx.x * blockDim.x + threadIdx.x;   // float4 index
    unsigned row  = v >> d4shift;                            // row in [0, M)
    unsigned col4 = v & ((1u << d4shift) - 1u);
    unsigned mi   = (unsigned)*mem_index;

    unsigned t = row + M - mi;                 // in [1, 2M)
    unsigned i = (t >= M) ? (t - M) : t;       // (row - mi) mod M, no idiv

    const v4f* src = (i < B)
        ? ((const v4f*)features + (((size_t)i)   << d4shift) + col4)
        : ((const v4f*)mem      + (((size_t)row) << d4shift) + col4);

    v4f val = __builtin_nontemporal_load(src);                 // global_load_b128 (NT)
    __builtin_nontemporal_store(val, (v4f*)out + (size_t)v);   // global_store_b128 (NT)
}

// ---------------------------------------------------------------------------
// Kernel B: new_util[r] = (r in window) ? q : util[r]; plus per-block partial
// sums of new_util[r] for r < new_index (feeds the mean reduction).
// ---------------------------------------------------------------------------
__global__ void util_kernel(const float* __restrict__ util_in,
                            const float* __restrict__ q_ptr,
                            const int* __restrict__ mem_index,
                            float* __restrict__ util_out,
                            float* __restrict__ partials,
                            unsigned M, unsigned B) {
    __shared__ float sdata[256];
    unsigned r  = blockIdx.x * blockDim.x + threadIdx.x;
    unsigned mi = (unsigned)*mem_index;
    float    q  = *q_ptr;

    unsigned t = r + M - mi;
    unsigned i = (t >= M) ? (t - M) : t;
    float val  = (i < B) ? q : util_in[r];
    util_out[r] = val;

    unsigned endw      = mi + B;
    unsigned new_index = (endw >= M) ? (endw - M) : endw;
    sdata[threadIdx.x] = (r < new_index) ? val : 0.0f;
    __syncthreads();

    for (unsigned s = blockDim.x >> 1; s > 0; s >>= 1) {
        if (threadIdx.x < s) sdata[threadIdx.x] += sdata[threadIdx.x + s];
        __syncthreads();
    }
    if (threadIdx.x == 0) partials[blockIdx.x] = sdata[0];
}

// ---------------------------------------------------------------------------
// Kernel C: single wave32. Reduce the block partials with
// V_WMMA_F32_16X16X4_F32 (A = ones 16x4, B = 64-partial chunk, accumulate in
// C). With ones-A:  D[m][n] = sum_k B[k][n] + C[m][n]  =>  sum(all of D) =
// 16 * (grand total), independent of the exact VGPR<->element mapping.
// Then a 32-lane shuffle reduce and the two scalar outputs.
// EXEC is all-ones (one full wave, no divergence) as WMMA requires.
// ---------------------------------------------------------------------------
__global__ void finalize_kernel(const float* __restrict__ partials,
                                const int* __restrict__ mem_index,
                                float* __restrict__ out_scalars, // [0]=utilization [1]=quality
                                unsigned numPartials, unsigned M, unsigned B) {
    unsigned lane = threadIdx.x;   // 0..31
    v2f a; a.x = 1.0f; a.y = 1.0f; // ones A-matrix (16x4 f32 = 2 VGPRs/lane)
    v8f c = {};                    // f32 accumulator (16x16 = 8 VGPRs/lane)

    for (unsigned base = 0; base < numPartials; base += 64) {
        v2f b;
        b.x = partials[base + lane];
        b.y = partials[base + 32u + lane];
        // v_wmma_f32_16x16x4_f32: D = A*B + C
        c = __builtin_amdgcn_wmma_f32_16x16x4_f32(
                /*neg_a=*/false, a, /*neg_b=*/false, b,
                /*c_mod=*/(short)0, c, /*reuse_a=*/false, /*reuse_b=*/false);
    }

    float s = c[0] + c[1] + c[2] + c[3] + c[4] + c[5] + c[6] + c[7];
    for (int off = 16; off > 0; off >>= 1)
        s += __shfl_xor(s, off, 32);

    if (lane == 0) {
        unsigned mi        = (unsigned)*mem_index;
        unsigned endw      = mi + B;
        bool     full      = (endw >= M);
        unsigned new_index = full ? (endw - M) : endw;
        float    total     = s * (1.0f / 16.0f);  // 16 identical D rows
        out_scalars[0] = full ? 1.0f : ((float)new_index / (float)M);
        out_scalars[1] = total / (float)new_index;
    }
}

// ---------------------------------------------------------------------------
// Launch: out layout = [new_mem (M*D) | new_util (M) | utilization | quality]
// ---------------------------------------------------------------------------
extern "C" void kernel_launch(void* const* d_in, const int* in_sizes, int n_in,
                              void* d_out, int out_size, void* d_ws, size_t ws_size,
                              hipStream_t stream) {
    const float* features  = (const float*)d_in[0];
    const float* quality   = (const float*)d_in[1];
    const float* mem       = (const float*)d_in[2];
    const float* util      = (const float*)d_in[3];
    const int*   mem_index = (const int*)d_in[4];

    const unsigned M  = (unsigned)in_sizes[3];            // 262144
    const unsigned MD = (unsigned)in_sizes[2];            // M * D
    const unsigned D  = MD / M;                           // 256
    const unsigned B  = (unsigned)in_sizes[0] / D;        // 16384
    const unsigned d4shift = (unsigned)__builtin_ctz(D / 4); // log2(D/4) = 6

    float* out         = (float*)d_out;
    float* out_util    = out + (size_t)MD;
    float* out_scalars = out_util + M;
    float* partials    = (float*)d_ws;                    // numPartials floats

    const unsigned threads = 256;
    const unsigned nVec    = MD / 4;                      // float4 count
    mem_copy_kernel<<<nVec / threads, threads, 0, stream>>>(
        features, mem, mem_index, out, M, B, d4shift);

    const unsigned nBlocksU = M / threads;                // 1024 (multiple of 64)
    util_kernel<<<nBlocksU, threads, 0, stream>>>(
        util, quality, mem_index, out_util, partials, M, B);

    finalize_kernel<<<1, 32, 0, stream>>>(
        partials, mem_index, out_scalars, nBlocksU, M, B);
}